// Count_37091337568592
// MI455X (gfx1250) — compile-verified
//
#include <hip/hip_runtime.h>

// Problem constants (fixed by the reference setup)
#define BB   8
#define CC   3
#define HH   768
#define WW   768
#define HWSZ (HH * WW)          // 589824
#define CHW  (CC * HWSZ)        // 1769472
#define NPIX (BB * HWSZ)        // 4718592
#define NQUAD (NPIX / 4)        // 1179648

typedef float v4f __attribute__((ext_vector_type(4)));

// Device-scope, non-returning hardware float atomic add (single L2 RMW,
// no CAS loop). STOREcnt-tracked; s_endpgm's implicit wait-idle drains it.
__device__ __forceinline__ void gatomic_add_f32(float* p, float v) {
  asm volatile("global_atomic_add_f32 %0, %1, off scope:SCOPE_DEV"
               :: "v"(p), "v"(v) : "memory");
}

__device__ __forceinline__ int wrap768(int v) {
  int m = v % 768;
  return m + ((m >> 31) & 768);   // add 768 iff negative
}

__global__ __launch_bounds__(256) void zero_fill_kernel(v4f* __restrict__ out, int n4) {
  int i = blockIdx.x * blockDim.x + threadIdx.x;
  if (i < n4) out[i] = (v4f){0.f, 0.f, 0.f, 0.f};
}

__global__ __launch_bounds__(256) void splat2d_kernel(const float* __restrict__ x,
                                                      const float* __restrict__ phi,
                                                      float* __restrict__ out) {
  int t = blockIdx.x * blockDim.x + threadIdx.x;
  if (t >= NQUAD) return;

  int p = t << 2;                 // first pixel of this quad
  int b = p / HWSZ;
  int r = p - b * HWSZ;           // h*W + w within the image
  int h = r / WW;
  int w = r - h * WW;             // W % 4 == 0 -> quad stays in one row

  const float* phib = phi + (size_t)b * 2 * HWSZ + r;
  const float* xb   = x   + (size_t)b * CHW + r;

  // Streaming inputs: 128-bit non-temporal loads (keep L2 for the atomics).
  v4f d_i = __builtin_nontemporal_load((const v4f*)(phib));           // phi[...,0]
  v4f d_j = __builtin_nontemporal_load((const v4f*)(phib + HWSZ));    // phi[...,1]
  v4f xc0 = __builtin_nontemporal_load((const v4f*)(xb));
  v4f xc1 = __builtin_nontemporal_load((const v4f*)(xb + HWSZ));
  v4f xc2 = __builtin_nontemporal_load((const v4f*)(xb + 2 * HWSZ));

  float* ob = out + (size_t)b * CHW;

#pragma unroll
  for (int j = 0; j < 4; ++j) {
    float gi = (float)h + d_i[j];
    float gj = (float)(w + j) + d_j[j];
    float i0f = floorf(gi);
    float j0f = floorf(gj);
    float fi = gi - i0f;
    float fj = gj - j0f;

    int i0 = wrap768((int)i0f);
    int j0 = wrap768((int)j0f);
    int i1 = i0 + 1; if (i1 == HH) i1 = 0;
    int j1 = j0 + 1; if (j1 == WW) j1 = 0;

    float w00 = (1.f - fi) * (1.f - fj);
    float w01 = (1.f - fi) * fj;
    float w10 = fi * (1.f - fj);
    float w11 = fi * fj;

    int a00 = i0 * WW + j0;
    int a01 = i0 * WW + j1;
    int a10 = i1 * WW + j0;
    int a11 = i1 * WW + j1;

    float xs0 = xc0[j], xs1 = xc1[j], xs2 = xc2[j];

    float* o0 = ob;
    float* o1 = ob + HWSZ;
    float* o2 = ob + 2 * HWSZ;

    gatomic_add_f32(o0 + a00, xs0 * w00);
    gatomic_add_f32(o0 + a01, xs0 * w01);
    gatomic_add_f32(o0 + a10, xs0 * w10);
    gatomic_add_f32(o0 + a11, xs0 * w11);

    gatomic_add_f32(o1 + a00, xs1 * w00);
    gatomic_add_f32(o1 + a01, xs1 * w01);
    gatomic_add_f32(o1 + a10, xs1 * w10);
    gatomic_add_f32(o1 + a11, xs1 * w11);

    gatomic_add_f32(o2 + a00, xs2 * w00);
    gatomic_add_f32(o2 + a01, xs2 * w01);
    gatomic_add_f32(o2 + a10, xs2 * w10);
    gatomic_add_f32(o2 + a11, xs2 * w11);
  }
}

extern "C" void kernel_launch(void* const* d_in, const int* in_sizes, int n_in,
                              void* d_out, int out_size, void* d_ws, size_t ws_size,
                              hipStream_t stream) {
  const float* x   = (const float*)d_in[0];   // [B,C,H,W] fp32
  const float* phi = (const float*)d_in[1];   // [B,2,H,W] fp32
  float* out = (float*)d_out;                 // [B,C,H,W] fp32

  // 1) Zero the accumulation surface (harness poisons d_out).
  int n4 = out_size / 4;                      // 3,538,944 float4s
  zero_fill_kernel<<<(n4 + 255) / 256, 256, 0, stream>>>((v4f*)out, n4);

  // 2) Bilinear scatter-add with hardware f32 atomics.
  splat2d_kernel<<<(NQUAD + 255) / 256, 256, 0, stream>>>(x, phi, out);
}